// ex_HT_76871324664526
// MI455X (gfx1250) — compile-verified
//
#include <hip/hip_runtime.h>

// Problem constants from the reference
#define BB 256
#define SS 168
#define MM 9
#define NPAIR (BB * SS)          // 43008 (b,s) pairs
#define NW 186                   // C + D weight rows
#define TPB 256
#define PAIRS_PER_BLOCK 256
#define OUT_PER_BLOCK (PAIRS_PER_BLOCK * MM)   // 2304

__global__ __launch_bounds__(TPB)
void ht_fused_kernel(const float* __restrict__ X,     // (B,S,M,1)
                     const float* __restrict__ Xex,   // (B,S,5)
                     const float* __restrict__ W,     // (186,1)
                     const float* __restrict__ bias,  // (1,)
                     float* __restrict__ out)         // (B,S,M,1)
{
    __shared__ float w_lds[NW];                       // 744 B, staged by TDM
    __shared__ float xex_lds[PAIRS_PER_BLOCK * 5];    // 5 KB, staged by async-to-LDS
    __shared__ float dot_lds[PAIRS_PER_BLOCK];

    const int t    = threadIdx.x;
    const int pair = blockIdx.x * PAIRS_PER_BLOCK + t;
    const long obase = (long)blockIdx.x * OUT_PER_BLOCK;

    // ---- Phase 0a: wave 0 programs the Tensor Data Mover to DMA W into LDS.
    //      D# group0: count=1 | lds_addr | global_addr | type=2
    //      D# group1: data_size=4B, tensor_dim0=tile_dim0=186, tile_dim1=1,
    //                 dim0_stride=186, workgroup_mask=0 (not in a cluster).
    if (t < 32) {
        const unsigned lds_w  = (unsigned)(uintptr_t)(&w_lds[0]);
        const unsigned long long ga = (unsigned long long)(uintptr_t)W;
        const unsigned ga_lo  = (unsigned)(ga & 0xffffffffu);
        const unsigned ga_hi  = (unsigned)((ga >> 32) & 0x01ffffffu) | 0x80000000u; // type=2
        asm volatile(
            "s_mov_b32 s20, 1\n\t"            // g0.w0: count=1
            "s_mov_b32 s21, %[ldsa]\n\t"      // g0.w1: lds_addr
            "s_mov_b32 s22, %[galo]\n\t"      // g0.w2: global_addr[31:0]
            "s_mov_b32 s23, %[gahi]\n\t"      // g0.w3: global_addr[56:32] | type<<30
            "s_mov_b32 s24, 0x20000\n\t"      // g1.w0: data_size=2 (4B), mask=0
            "s_mov_b32 s25, 0xBA0000\n\t"     // g1.w1: tensor_dim0.lo16=186 << 16
            "s_mov_b32 s26, 0x10000\n\t"      // g1.w2: tensor_dim1.lo16=1  << 16
            "s_mov_b32 s27, 0xBA0000\n\t"     // g1.w3: tile_dim0=186 << 16
            "s_mov_b32 s28, 1\n\t"            // g1.w4: tile_dim1=1, tile_dim2=0
            "s_mov_b32 s29, 186\n\t"          // g1.w5: tensor_dim0_stride=186
            "s_mov_b32 s30, 0\n\t"            // g1.w6
            "s_mov_b32 s31, 0\n\t"            // g1.w7
            "tensor_load_to_lds s[20:23], s[24:31]\n\t"
            :
            : [ldsa] "s"(lds_w), [galo] "s"(ga_lo), [gahi] "s"(ga_hi)
            : "s20","s21","s22","s23","s24","s25","s26","s27","s28","s29","s30","s31",
              "memory");
    }

    // ---- Phase 0b: every lane stages its 5-float X_ex row into LDS via the
    //      CDNA5 async global->LDS pipe (ASYNCcnt), overlapped with the TDM. ----
    {
        const float* gsrc = Xex + (long)pair * 5;
        #pragma unroll
        for (int j = 0; j < 5; ++j) {
            unsigned lds_off = (unsigned)(uintptr_t)(&xex_lds[t * 5 + j]);
            const float* g = gsrc + j;
            asm volatile("global_load_async_to_lds_b32 %0, %1, off"
                         :: "v"(lds_off), "v"(g) : "memory");
        }
    }

    // Prefetch this block's 9216B X span while the DMAs fly.
    __builtin_prefetch(&X[obase + (long)t * MM], 0, 0);

    asm volatile("s_wait_asynccnt 0" ::: "memory");
    if (t < 32) {
        __builtin_amdgcn_s_wait_tensorcnt(0);
    }
    __syncthreads();

    // ---- Phase 1: one thread per (b,s) pair -> 13-nonzero sparse dot ----
    {
        const float* xr = &xex_lds[t * 5];
        const float f_day = xr[0], f_hr = xr[1], f_mo = xr[2], f_wd = xr[3];
        const float t1 = xr[4];
        const int day   = (int)f_day - 1;   // 0..30
        const int hour  = (int)f_hr;        // 0..23
        const int month = (int)f_mo - 1;    // 0..11
        const int wday  = (int)f_wd;        // 0..6
        const float t2 = t1 * t1;
        const float t3 = t2 * t1;

        const float* w1 = w_lds + 1;        // skip W[0] (the X column weight)
        float d = w1[hour] + w1[24 + day] + w1[55 + wday] + w1[62 + month];
        d = fmaf(t1, w1[74], d);
        d = fmaf(t2, w1[75], d);
        d = fmaf(t3, w1[76], d);
        d = fmaf(t1, w1[77  + hour],  d);
        d = fmaf(t2, w1[101 + hour],  d);
        d = fmaf(t3, w1[125 + hour],  d);
        d = fmaf(t1, w1[149 + month], d);
        d = fmaf(t2, w1[161 + month], d);
        d = fmaf(t3, w1[173 + month], d);
        dot_lds[t] = d + bias[0];
    }
    __syncthreads();

    // ---- Phase 2: coalesced sweep of the block's 2304 outputs ----
    const float scale = w_lds[0] + 1.0f;    // W[0,0] + residual add of X
    #pragma unroll
    for (int i = 0; i < MM; ++i) {
        const int idx = i * TPB + t;        // 0..2303
        const long g  = obase + idx;
        out[g] = fmaf(X[g], scale, dot_lds[idx / MM]);
    }
}

extern "C" void kernel_launch(void* const* d_in, const int* in_sizes, int n_in,
                              void* d_out, int out_size, void* d_ws, size_t ws_size,
                              hipStream_t stream) {
    const float* X    = (const float*)d_in[0];
    const float* Xex  = (const float*)d_in[1];
    const float* W    = (const float*)d_in[2];
    const float* bias = (const float*)d_in[3];
    float* out        = (float*)d_out;

    dim3 grid(NPAIR / PAIRS_PER_BLOCK);     // 168 blocks
    dim3 block(TPB);                        // 8 wave32s / block
    ht_fused_kernel<<<grid, block, 0, stream>>>(X, Xex, W, bias, out);
}